// LSTM_AE_79018808312411
// MI455X (gfx1250) — compile-verified
//
#include <hip/hip_runtime.h>
#include <hip/hip_bf16.h>
#include <math.h>

// ---------------------------------------------------------------------------
// LSTM autoencoder, single timestep per cell, h0 = c0 = 0.
// Every stage is an fp32 matvec (HBM bandwidth bound on MI455X, 23.3 TB/s).
//   * whh never read (h0 = 0); f-gate rows of wih never read (c0 = 0)
//     -> total traffic ~260 MB, ~11 us roofline.
//   * dot products via V_WMMA_F32_16X16X4_F32 (exact fp32 matrix op):
//     A = 16 weight rows x 4 K, B = x broadcast over all 16 N columns.
//   * b128 weight loads: one float4 per lane covers two WMMAs (K relabeling
//     inside a WMMA is legal since we build B to match slot-for-slot).
//   * x chunk staged in LDS once per block, read via ds_load_b128 so the
//     vmem pipe carries only the weight stream.
//   * K split across blocks; partials reduced deterministically (no atomics).
// ---------------------------------------------------------------------------

typedef __attribute__((ext_vector_type(2))) float v2f;
typedef __attribute__((ext_vector_type(4))) float v4f;
typedef __attribute__((ext_vector_type(8))) float v8f;

#define KSPLIT 4
#define WPB 8                   // waves per block (256 threads, wave32)
#define MAX_KC 1024             // max din/KSPLIT = 4096/4

// mode 0: LSTM 3-gate row remap (logical rows [0,3*dh) -> wih rows i/g/o)
// mode 1: identity (plain linear matvec, ntiles*16 rows)
// Grid: (ntiles/WPB) * KSPLIT blocks; all waves of a block share one K-chunk.
__global__ __launch_bounds__(256) void wmma_matvec_partial(
    const float* __restrict__ W, const float* __restrict__ x,
    float* __restrict__ partial, int din, int dh, int mode, int ntiles)
{
    __shared__ float xs[MAX_KC];

    const int lane = threadIdx.x & 31;
    const int wave = threadIdx.x >> 5;

    const int Kc        = din / KSPLIT;           // multiple of 256
    const int ks        = blockIdx.x % KSPLIT;
    const int tileBlock = blockIdx.x / KSPLIT;
    const int tile      = tileBlock * WPB + wave; // ntiles % WPB == 0 by construction
    const int k0        = ks * Kc;

    // Stage x[k0 .. k0+Kc) into LDS (<= 4 KB), float4 per thread.
    for (int i = threadIdx.x; i < (Kc >> 2); i += 256) {
        *(v4f*)(&xs[i << 2]) = *(const v4f*)(x + k0 + (i << 2));
    }
    __syncthreads();

    const int r = tile * 16;                      // logical (gate-relative) row base
    int wrow_base;
    if (mode == 0) {
        const int gate = r / dh;                  // 0,1,2 -> i,g,o
        const int j    = r - gate * dh;
        const int wsel = (gate == 0) ? 0 : (gate + 1);   // i=0, g=2, o=3 (skip f=1)
        wrow_base = wsel * dh + j;
    } else {
        wrow_base = r;
    }

    // 16x4 fp32 A layout: lane L -> row L%16, K slots 2*(L/16)+{0,1}.
    // We relabel K: lane reads W[row, k + 4h .. k+4h+3] (h = L/16) as b128;
    // halves feed two WMMAs whose B operands use identical x indices.
    const int m  = lane & 15;
    const int hh = (lane >> 4) << 2;              // 0 or 4
    const float* __restrict__ wp = W + (size_t)(wrow_base + m) * din + k0 + hh;
    const float* __restrict__ bp = xs + hh;

    v8f acc = {0.f, 0.f, 0.f, 0.f, 0.f, 0.f, 0.f, 0.f};
    #pragma unroll 4
    for (int k = 0; k < Kc; k += 8) {
        v4f a4 = *(const v4f*)(wp + k);           // global_load_b128: weight stream
        v4f b4 = *(const v4f*)(bp + k);           // ds_load_b128: shared vector
        v2f alo = __builtin_shufflevector(a4, a4, 0, 1);
        v2f ahi = __builtin_shufflevector(a4, a4, 2, 3);
        v2f blo = __builtin_shufflevector(b4, b4, 0, 1);
        v2f bhi = __builtin_shufflevector(b4, b4, 2, 3);
        acc = __builtin_amdgcn_wmma_f32_16x16x4_f32(
                  false, alo, false, blo, (short)0, acc, false, false);
        acc = __builtin_amdgcn_wmma_f32_16x16x4_f32(
                  false, ahi, false, bhi, (short)0, acc, false, false);
    }

    // D layout: VGPR v, lanes 0-15 hold M=v, lanes 16-31 hold M=v+8.
    // All N columns identical (B broadcast) -> lanes 0 and 16 write 8 rows each.
    float* __restrict__ out = partial + (size_t)ks * (ntiles * 16) + r;
    if (lane == 0) {
        #pragma unroll
        for (int v = 0; v < 8; ++v) out[v] = acc[v];
    } else if (lane == 16) {
        #pragma unroll
        for (int v = 0; v < 8; ++v) out[8 + v] = acc[v];
    }
}

__device__ __forceinline__ float sigmoidf_(float v) {
    return 1.f / (1.f + __expf(-v));
}

// Reduce K-split partials, add biases, apply LSTM nonlinearity (f*c0 = 0).
__global__ __launch_bounds__(256) void lstm_activation(
    const float* __restrict__ partial, const float* __restrict__ bih,
    const float* __restrict__ bhh, float* __restrict__ h, int dh)
{
    const int j = blockIdx.x * blockDim.x + threadIdx.x;
    if (j >= dh) return;
    const int stride = 3 * dh;
    float si = 0.f, sg = 0.f, so = 0.f;
    #pragma unroll
    for (int ks = 0; ks < KSPLIT; ++ks) {
        const float* p = partial + (size_t)ks * stride;
        si += p[j];
        sg += p[dh + j];
        so += p[2 * dh + j];
    }
    si += bih[j]          + bhh[j];            // i-gate rows [0,dh)
    sg += bih[2 * dh + j] + bhh[2 * dh + j];   // g-gate rows [2dh,3dh)
    so += bih[3 * dh + j] + bhh[3 * dh + j];   // o-gate rows [3dh,4dh)
    const float c = sigmoidf_(si) * tanhf(sg);
    h[j] = sigmoidf_(so) * tanhf(c);
}

// Final linear: reduce partials + bias.
__global__ __launch_bounds__(256) void linear_finalize(
    const float* __restrict__ partial, const float* __restrict__ bias,
    float* __restrict__ y, int n)
{
    const int j = blockIdx.x * blockDim.x + threadIdx.x;
    if (j >= n) return;
    float s = bias[j];
    #pragma unroll
    for (int ks = 0; ks < KSPLIT; ++ks) s += partial[(size_t)ks * n + j];
    y[j] = s;
}

static inline void launch_cell(const float* wih, const float* bih, const float* bhh,
                               const float* x, float* hout, float* partial,
                               int din, int dh, hipStream_t stream)
{
    const int ntiles = (3 * dh) / 16;            // 192 or 384: divisible by WPB
    const int blocks = (ntiles / WPB) * KSPLIT;
    wmma_matvec_partial<<<blocks, 256, 0, stream>>>(wih, x, partial, din, dh, 0, ntiles);
    lstm_activation<<<(dh + 255) / 256, 256, 0, stream>>>(partial, bih, bhh, hout, dh);
}

extern "C" void kernel_launch(void* const* d_in, const int* in_sizes, int n_in,
                              void* d_out, int out_size, void* d_ws, size_t ws_size,
                              hipStream_t stream)
{
    (void)in_sizes; (void)n_in; (void)out_size; (void)ws_size;
    const int F = 4096, E = 1024, H = 2048;

    const float* x = (const float*)d_in[0];
    // Per-cell params: _wih, _whh(unused), _bih, _bhh
    const float* e1l0_wih = (const float*)d_in[1];
    const float* e1l0_bih = (const float*)d_in[3];
    const float* e1l0_bhh = (const float*)d_in[4];
    const float* e1l1_wih = (const float*)d_in[5];
    const float* e1l1_bih = (const float*)d_in[7];
    const float* e1l1_bhh = (const float*)d_in[8];
    const float* e2_wih   = (const float*)d_in[9];
    const float* e2_bih   = (const float*)d_in[11];
    const float* e2_bhh   = (const float*)d_in[12];
    const float* d1l0_wih = (const float*)d_in[13];
    const float* d1l0_bih = (const float*)d_in[15];
    const float* d1l0_bhh = (const float*)d_in[16];
    const float* d1l1_wih = (const float*)d_in[17];
    const float* d1l1_bih = (const float*)d_in[19];
    const float* d1l1_bhh = (const float*)d_in[20];
    const float* d2_wih   = (const float*)d_in[21];
    const float* d2_bih   = (const float*)d_in[23];
    const float* d2_bhh   = (const float*)d_in[24];
    const float* out_w    = (const float*)d_in[25];
    const float* out_b    = (const float*)d_in[26];

    // Workspace layout (floats):
    //   [0, 32768)    : K-split partials (max KSPLIT*3*H = 24576)
    //   [32768, ...)  : intermediate vectors h1,h2,z,h3,h4,h5
    float* ws      = (float*)d_ws;
    float* partial = ws;
    float* vecs    = ws + 32768;
    float* h1 = vecs;            // H
    float* h2 = h1 + H;          // H
    float* z  = h2 + H;          // E
    float* h3 = z  + E;          // E
    float* h4 = h3 + E;          // E
    float* h5 = h4 + E;          // H

    // Encoder
    launch_cell(e1l0_wih, e1l0_bih, e1l0_bhh, x,  h1, partial, F, H, stream);
    launch_cell(e1l1_wih, e1l1_bih, e1l1_bhh, h1, h2, partial, H, H, stream);
    launch_cell(e2_wih,   e2_bih,   e2_bhh,   h2, z,  partial, H, E, stream);
    // Decoder
    launch_cell(d1l0_wih, d1l0_bih, d1l0_bhh, z,  h3, partial, E, E, stream);
    launch_cell(d1l1_wih, d1l1_bih, d1l1_bhh, h3, h4, partial, E, E, stream);
    launch_cell(d2_wih,   d2_bih,   d2_bhh,   h4, h5, partial, E, H, stream);
    // Output linear: out_w (F x H) @ h5 + out_b
    {
        const int ntiles = F / 16;               // 256: divisible by WPB
        const int blocks = (ntiles / WPB) * KSPLIT;
        wmma_matvec_partial<<<blocks, 256, 0, stream>>>(out_w, h5, partial, H, F, 1, ntiles);
        linear_finalize<<<(F + 255) / 256, 256, 0, stream>>>(partial, out_b, (float*)d_out, F);
    }
}